// TMSpectralConv1D_38439957299538
// MI455X (gfx1250) — compile-verified
//
#include <hip/hip_runtime.h>
#include <hip/hip_bf16.h>
#include <math.h>

// ---------------------------------------------------------------------------
// TMSpectralConv1D for MI455X (gfx1250), wave32, f32 WMMA 16x16x4 + TDM.
//
//   K0 fill_tables : forward DFT matrix F (272x2048, zero-padded rows) and
//                    weighted synthesis matrix G (2048x272) in d_ws.
//   K0b zero_ypad  : zero Y rows k=257..271 so padded contraction is exact.
//   K1 tmod        : t_tr[b,k] = t_emb @ tW_real + i * t_emb @ tW_imag.
//   K2 fwd_dft     : X[k,b,i] = sum_n F[k,n] x[b,n,i]   (WMMA, A via TDM->LDS)
//   K3 mode_gemm   : Y[b,k,o] = (X_k @ W_k) * t_tr[b,k] (WMMA, complex)
//   K4 inv_dft     : out[b,n,o] = sum_k GrYr + GiYi     (WMMA, A via TDM->LDS)
// ---------------------------------------------------------------------------

typedef __attribute__((ext_vector_type(2))) float        v2f;
typedef __attribute__((ext_vector_type(8))) float        v8f;
typedef __attribute__((ext_vector_type(4))) unsigned int v4u;
typedef __attribute__((ext_vector_type(8))) int          v8i;
typedef __attribute__((ext_vector_type(4))) int          v4i;

#define BB    128      // batch
#define NN    2048     // sequence length
#define DIN   128
#define DOUT  128
#define ENC   256
#define MM    257      // retained modes
#define MP    272      // modes padded to 17 tiles of 16

// workspace layout (float offsets)
#define SZ_F  ((size_t)MP * NN)          // 557056
#define SZ_X  ((size_t)MP * BB * DIN)    // 4456448
#define SZ_Y  ((size_t)BB * MP * DOUT)   // 4456448
#define SZ_T  ((size_t)BB * MP)          // 34816
#define OFF_FR ((size_t)0)
#define OFF_FI (OFF_FR + SZ_F)
#define OFF_GR (OFF_FI + SZ_F)
#define OFF_GI (OFF_GR + SZ_F)
#define OFF_XR (OFF_GI + SZ_F)
#define OFF_XI (OFF_XR + SZ_X)
#define OFF_YR (OFF_XI + SZ_X)
#define OFF_YI (OFF_YR + SZ_Y)
#define OFF_TR (OFF_YI + SZ_Y)
#define OFF_TI (OFF_TR + SZ_T)

__device__ __forceinline__ v8f wmma_f32(v2f a, v2f b, v8f c) {
  // D = A(16x4 f32) x B(4x16 f32) + C(16x16 f32)
  return __builtin_amdgcn_wmma_f32_16x16x4_f32(false, a, false, b, (short)0, c,
                                               false, false);
}

// --- TDM: load a 2-D tile of f32 (tile_w x tile_h) from a row-major tensor
//     into LDS at lds_off (bytes). Descriptor per CDNA5 ISA ch.8.
//     pad_*: optional LDS row padding (pad every (2<<pad_interval_code) DW by
//     (pad_amount_code+1) DW) for bank-conflict-free reads.
__device__ __forceinline__ void tdm_load_2d(unsigned lds_off,
                                            const float* gptr,
                                            int tile_w, int tile_h,
                                            int tensor_w, int tensor_h,
                                            int row_stride_elems,
                                            int pad_enable,
                                            int pad_interval_code,
                                            int pad_amount_code) {
  const unsigned long long ga = (unsigned long long)(uintptr_t)gptr;
  v4u g0;
  g0[0] = 1u;                                   // count=1, user descriptor
  g0[1] = lds_off;                              // lds_addr [63:32]
  g0[2] = (unsigned)(ga & 0xffffffffu);         // global_addr [95:64]
  g0[3] = (unsigned)((ga >> 32) & 0x01ffffffu)  // global_addr [120:96]
          | (2u << 30);                         // type=2 ("image")
  v8i g1;
  g1[0] = (int)((2u << 16)                          // data_size = 4B
                | ((unsigned)pad_enable << 20)
                | ((unsigned)pad_interval_code << 22)
                | ((unsigned)pad_amount_code << 25));
  g1[1] = (int)(((unsigned)tensor_w & 0xffffu) << 16);   // tensor_dim0 lo
  g1[2] = (int)((((unsigned)tensor_w >> 16) & 0xffffu)   // tensor_dim0 hi
                | (((unsigned)tensor_h & 0xffffu) << 16));// tensor_dim1 lo
  g1[3] = (int)((((unsigned)tensor_h >> 16) & 0xffffu)   // tensor_dim1 hi
                | ((unsigned)tile_w << 16));             // tile_dim0
  g1[4] = (int)((unsigned)tile_h & 0xffffu);             // tile_dim1 (dim2=0)
  g1[5] = (int)(unsigned)row_stride_elems;               // dim0_stride lo32
  g1[6] = 0;                                             // stride hi / dim1_stride
  g1[7] = 0;
  const v4i z4 = {0, 0, 0, 0};                 // 2-D: groups 2/3 NULL
  const v8i z8 = {0, 0, 0, 0, 0, 0, 0, 0};     // unused extra group (6-arg form)
  __builtin_amdgcn_tensor_load_to_lds(g0, g1, z4, z4, z8, 0);
}

// --- K0: exact twiddle tables via (k*n) mod 2048 reduction -----------------
__global__ __launch_bounds__(256) void fill_tables(float* __restrict__ Fr,
                                                   float* __restrict__ Fi,
                                                   float* __restrict__ Gr,
                                                   float* __restrict__ Gi) {
  const int idx = blockIdx.x * 256 + threadIdx.x;
  if (idx >= (int)(MP * NN)) return;
  const float step = 6.28318530717958647692f / (float)NN;
  {  // F: [k][n] row-major, rows k>=257 are zero
    const int k = idx / NN, n = idx % NN;
    float fr = 0.f, fi = 0.f;
    if (k < MM) {
      const int r = (k * n) & (NN - 1);
      float s, c;
      sincosf((float)r * step, &s, &c);
      fr = c;
      fi = -s;
    }
    Fr[idx] = fr;
    Fi[idx] = fi;
  }
  {  // G: [n][k] row-major (idx = n*MP + k), cols k>=257 are zero
    const int n = idx / MP, k = idx % MP;
    float gr = 0.f, gi = 0.f;
    if (k < MM) {
      const int r = (k * n) & (NN - 1);
      float s, c;
      sincosf((float)r * step, &s, &c);
      const float w = (k == 0) ? (1.0f / (float)NN) : (2.0f / (float)NN);
      gr = w * c;
      gi = -w * s;
    }
    Gr[idx] = gr;
    Gi[idx] = gi;
  }
}

// --- K0b: zero padded Y rows (k = 257..271) --------------------------------
__global__ __launch_bounds__(256) void zero_ypad(float* __restrict__ Yr,
                                                 float* __restrict__ Yi) {
  const int idx = blockIdx.x * 256 + threadIdx.x;
  const int total = BB * (MP - MM) * DOUT;
  if (idx >= total) return;
  const int b = idx / ((MP - MM) * DOUT);
  const int rem = idx % ((MP - MM) * DOUT);
  const int k = MM + rem / DOUT;
  const int o = rem % DOUT;
  const size_t off = (size_t)b * MP * DOUT + (size_t)k * DOUT + o;
  Yr[off] = 0.f;
  Yi[off] = 0.f;
}

// --- K1: t_tr = t_emb @ tW_real + i * t_emb @ tW_imag ----------------------
__global__ __launch_bounds__(256) void tmod(const float* __restrict__ t_emb,
                                            const float* __restrict__ tWr,
                                            const float* __restrict__ tWi,
                                            float* __restrict__ Tr,
                                            float* __restrict__ Ti) {
  const int idx = blockIdx.x * 256 + threadIdx.x;
  if (idx >= BB * MP) return;
  const int b = idx / MP, k = idx % MP;
  float ar = 0.f, ai = 0.f;
  if (k < MM) {
    const float* te = t_emb + (size_t)b * ENC;
#pragma unroll 4
    for (int e = 0; e < ENC; ++e) {
      const float t = te[e];
      ar = fmaf(t, tWr[(size_t)e * MM + k], ar);
      ai = fmaf(t, tWi[(size_t)e * MM + k], ai);
    }
  }
  Tr[idx] = ar;
  Ti[idx] = ai;
}

// --- K2: forward DFT GEMM with TDM-staged A tiles --------------------------
// grid(17 ktiles, 128 b), block 256 (8 waves, one 16-wide i-tile per wave).
// A (F rows) staged 64 n at a time into LDS (row pitch 66 via TDM padding),
// double-buffered, shared by all 8 waves.
#define CH2   64                   // n per stage
#define NST2  (NN / CH2)           // 32 stages
#define PITCH2 66                  // 64 DW + 2 DW TDM pad
__global__ __launch_bounds__(256) void fwd_dft(const float* __restrict__ x,
                                               const float* __restrict__ Fr,
                                               const float* __restrict__ Fi,
                                               float* __restrict__ Xr,
                                               float* __restrict__ Xi) {
  __shared__ float sFr[2][16 * PITCH2];
  __shared__ float sFi[2][16 * PITCH2];
  const int lane = threadIdx.x & 31;
  const int wave = threadIdx.x >> 5;
  const int half = lane >> 4;
  const int col  = lane & 15;
  const int k0 = blockIdx.x * 16;
  const int b  = blockIdx.y;
  const int i0 = wave * 16;
  const float* __restrict__ xb = x + (size_t)b * NN * DIN;
  const float* fr0 = Fr + (size_t)k0 * NN;
  const float* fi0 = Fi + (size_t)k0 * NN;

  if (wave == 0) {  // prologue: stage 0 into buffer 0
    tdm_load_2d((unsigned)(uintptr_t)&sFr[0][0], fr0, CH2, 16, NN, MP, NN,
                1, 5, 1);
    tdm_load_2d((unsigned)(uintptr_t)&sFi[0][0], fi0, CH2, 16, NN, MP, NN,
                1, 5, 1);
  }

  v8f accR = {};
  v8f accI = {};
  for (int s = 0; s < NST2; ++s) {
    const int buf = s & 1;
    if (wave == 0) {
      if (s + 1 < NST2) {  // prefetch next stage, then wait for current
        tdm_load_2d((unsigned)(uintptr_t)&sFr[buf ^ 1][0],
                    fr0 + (size_t)(s + 1) * CH2, CH2, 16, NN, MP, NN, 1, 5, 1);
        tdm_load_2d((unsigned)(uintptr_t)&sFi[buf ^ 1][0],
                    fi0 + (size_t)(s + 1) * CH2, CH2, 16, NN, MP, NN, 1, 5, 1);
        __builtin_amdgcn_s_wait_tensorcnt(2);
      } else {
        __builtin_amdgcn_s_wait_tensorcnt(0);
      }
    }
    __syncthreads();
#pragma unroll
    for (int t = 0; t < CH2; t += 4) {
      const int ta = t + 2 * half;
      const v2f ar = *(const v2f*)(&sFr[buf][col * PITCH2 + ta]);
      const v2f ai = *(const v2f*)(&sFi[buf][col * PITCH2 + ta]);
      const int n = s * CH2 + ta;
      v2f bv;
      bv.x = xb[(size_t)n * DIN + i0 + col];
      bv.y = xb[(size_t)(n + 1) * DIN + i0 + col];
      accR = wmma_f32(ar, bv, accR);
      accI = wmma_f32(ai, bv, accI);
    }
    __syncthreads();
  }
#pragma unroll
  for (int v = 0; v < 8; ++v) {
    const int k = k0 + v + 8 * half;
    const size_t off = (size_t)k * BB * DIN + (size_t)b * DIN + i0 + col;
    Xr[off] = accR[v];
    Xi[off] = accI[v];
  }
}

// --- K3: per-mode complex GEMM + time modulation ---------------------------
// grid(257 modes, 8), block 256; 64 tiles of (16b x 16o) per mode.
__global__ __launch_bounds__(256) void mode_gemm(const float* __restrict__ Xr,
                                                 const float* __restrict__ Xi,
                                                 const float* __restrict__ Wr,
                                                 const float* __restrict__ Wi,
                                                 const float* __restrict__ Tr,
                                                 const float* __restrict__ Ti,
                                                 float* __restrict__ Yr,
                                                 float* __restrict__ Yi) {
  const int lane = threadIdx.x & 31;
  const int wave = threadIdx.x >> 5;
  const int half = lane >> 4;
  const int col  = lane & 15;
  const int k = blockIdx.x;                  // 0..256
  const int tile = blockIdx.y * 8 + wave;    // 0..63
  const int b0 = (tile >> 3) * 16;
  const int o0 = (tile & 7) * 16;
  const float* __restrict__ xr = Xr + (size_t)k * BB * DIN;
  const float* __restrict__ xi = Xi + (size_t)k * BB * DIN;
  const float* __restrict__ wr = Wr + (size_t)k * DIN * DOUT;
  const float* __restrict__ wi = Wi + (size_t)k * DIN * DOUT;
  const size_t arow = (size_t)(b0 + col) * DIN;
  v8f aRR = {};   // sum Xr*Wr
  v8f aII = {};   // sum Xi*Wi (f32 WMMA has no A/B negate -> subtract later)
  v8f accI = {};  // sum Xr*Wi + Xi*Wr (chained accumulation)
#pragma unroll 4
  for (int i0 = 0; i0 < DIN; i0 += 4) {
    const int ia = i0 + 2 * half;
    const v2f axr = *(const v2f*)(xr + arow + ia);
    const v2f axi = *(const v2f*)(xi + arow + ia);
    v2f bwr, bwi;
    bwr.x = wr[(size_t)ia * DOUT + o0 + col];
    bwr.y = wr[(size_t)(ia + 1) * DOUT + o0 + col];
    bwi.x = wi[(size_t)ia * DOUT + o0 + col];
    bwi.y = wi[(size_t)(ia + 1) * DOUT + o0 + col];
    aRR  = wmma_f32(axr, bwr, aRR);
    aII  = wmma_f32(axi, bwi, aII);
    accI = wmma_f32(axr, bwi, accI);
    accI = wmma_f32(axi, bwr, accI);
  }
#pragma unroll
  for (int v = 0; v < 8; ++v) {
    const int b = b0 + v + 8 * half;
    const float yr = aRR[v] - aII[v];
    const float yi = accI[v];
    const float tr = Tr[(size_t)b * MP + k];
    const float ti = Ti[(size_t)b * MP + k];
    const size_t off = (size_t)b * MP * DOUT + (size_t)k * DOUT + o0 + col;
    Yr[off] = yr * tr - yi * ti;
    Yi[off] = yr * ti + yi * tr;
  }
}

// --- K4: inverse synthesis GEMM with TDM-staged A tiles --------------------
// grid(128 ntiles, 128 b), block 256 (wave = 16-wide o-tile).
// A (G rows) staged 68 k at a time (272 = 4*68; pitch 68 -> rows stride 4
// banks, naturally conflict-free), double-buffered.
#define CH4   68
#define NST4  (MP / CH4)           // 4 stages
__global__ __launch_bounds__(256) void inv_dft(const float* __restrict__ Gr,
                                               const float* __restrict__ Gi,
                                               const float* __restrict__ Yr,
                                               const float* __restrict__ Yi,
                                               float* __restrict__ out) {
  __shared__ float sGr[2][16 * CH4];
  __shared__ float sGi[2][16 * CH4];
  const int lane = threadIdx.x & 31;
  const int wave = threadIdx.x >> 5;
  const int half = lane >> 4;
  const int col  = lane & 15;
  const int n0 = blockIdx.x * 16;
  const int b  = blockIdx.y;
  const int o0 = wave * 16;
  const float* __restrict__ yrb = Yr + (size_t)b * MP * DOUT;
  const float* __restrict__ yib = Yi + (size_t)b * MP * DOUT;
  const float* gr0 = Gr + (size_t)n0 * MP;
  const float* gi0 = Gi + (size_t)n0 * MP;

  if (wave == 0) {
    tdm_load_2d((unsigned)(uintptr_t)&sGr[0][0], gr0, CH4, 16, MP, NN, MP,
                0, 0, 0);
    tdm_load_2d((unsigned)(uintptr_t)&sGi[0][0], gi0, CH4, 16, MP, NN, MP,
                0, 0, 0);
  }

  v8f acc = {};
  for (int s = 0; s < NST4; ++s) {
    const int buf = s & 1;
    if (wave == 0) {
      if (s + 1 < NST4) {
        tdm_load_2d((unsigned)(uintptr_t)&sGr[buf ^ 1][0],
                    gr0 + (size_t)(s + 1) * CH4, CH4, 16, MP, NN, MP, 0, 0, 0);
        tdm_load_2d((unsigned)(uintptr_t)&sGi[buf ^ 1][0],
                    gi0 + (size_t)(s + 1) * CH4, CH4, 16, MP, NN, MP, 0, 0, 0);
        __builtin_amdgcn_s_wait_tensorcnt(2);
      } else {
        __builtin_amdgcn_s_wait_tensorcnt(0);
      }
    }
    __syncthreads();
#pragma unroll
    for (int t = 0; t < CH4; t += 4) {
      const int ta = t + 2 * half;
      const v2f ac = *(const v2f*)(&sGr[buf][col * CH4 + ta]);
      const v2f as = *(const v2f*)(&sGi[buf][col * CH4 + ta]);
      const int ka = s * CH4 + ta;
      v2f br, bi;
      br.x = yrb[(size_t)ka * DOUT + o0 + col];
      br.y = yrb[(size_t)(ka + 1) * DOUT + o0 + col];
      bi.x = yib[(size_t)ka * DOUT + o0 + col];
      bi.y = yib[(size_t)(ka + 1) * DOUT + o0 + col];
      acc = wmma_f32(ac, br, acc);
      acc = wmma_f32(as, bi, acc);
    }
    __syncthreads();
  }
#pragma unroll
  for (int v = 0; v < 8; ++v) {
    const int n = n0 + v + 8 * half;
    out[(size_t)b * NN * DOUT + (size_t)n * DOUT + o0 + col] = acc[v];
  }
}

extern "C" void kernel_launch(void* const* d_in, const int* in_sizes, int n_in,
                              void* d_out, int out_size, void* d_ws,
                              size_t ws_size, hipStream_t stream) {
  const float* x      = (const float*)d_in[0];  // (B, N, DIN)
  const float* t_emb  = (const float*)d_in[1];  // (B, ENC)
  const float* W_real = (const float*)d_in[2];  // (M, DIN, DOUT)
  const float* W_imag = (const float*)d_in[3];
  const float* tW_r   = (const float*)d_in[4];  // (ENC, M)
  const float* tW_i   = (const float*)d_in[5];
  float* out = (float*)d_out;                   // (B, N, DOUT)

  float* ws = (float*)d_ws;
  float* Fr = ws + OFF_FR;
  float* Fi = ws + OFF_FI;
  float* Gr = ws + OFF_GR;
  float* Gi = ws + OFF_GI;
  float* Xr = ws + OFF_XR;
  float* Xi = ws + OFF_XI;
  float* Yr = ws + OFF_YR;
  float* Yi = ws + OFF_YI;
  float* Tr = ws + OFF_TR;
  float* Ti = ws + OFF_TI;

  {
    const int total = (int)(MP * NN);
    fill_tables<<<(total + 255) / 256, 256, 0, stream>>>(Fr, Fi, Gr, Gi);
  }
  {
    const int total = BB * (MP - MM) * DOUT;
    zero_ypad<<<(total + 255) / 256, 256, 0, stream>>>(Yr, Yi);
  }
  {
    const int total = BB * MP;
    tmod<<<(total + 255) / 256, 256, 0, stream>>>(t_emb, tW_r, tW_i, Tr, Ti);
  }
  fwd_dft<<<dim3(MP / 16, BB), 256, 0, stream>>>(x, Fr, Fi, Xr, Xi);
  mode_gemm<<<dim3(MM, 8), 256, 0, stream>>>(Xr, Xi, W_real, W_imag, Tr, Ti,
                                             Yr, Yi);
  inv_dft<<<dim3(NN / 16, BB), 256, 0, stream>>>(Gr, Gi, Yr, Yi, out);
}